// Block_81088982549089
// MI455X (gfx1250) — compile-verified
//
#include <hip/hip_runtime.h>
#include <math.h>

typedef float v2f __attribute__((ext_vector_type(2)));
typedef float v4f __attribute__((ext_vector_type(4)));
typedef float v8f __attribute__((ext_vector_type(8)));

#define EPSV 1e-5f
#define WAVES 2          // independent blocks (waves) per workgroup
#define CS 68            // column stride (dwords) for 64-row CM matrices
#define SS 20            // column stride for 16-row attention strip
#define PW 5632          // floats of LDS per wave

static __device__ __forceinline__ v8f wmma4(v2f a, v2f b, v8f c) {
  // D = A(16x4) * B(4x16) + C, all f32
  return __builtin_amdgcn_wmma_f32_16x16x4_f32(false, a, false, b, (short)0, c,
                                               false, false);
}

static __device__ __forceinline__ v2f ldpair(const float* m, int idx, int step) {
  v2f a;
  a[0] = m[idx];
  a[1] = m[idx + step];
  return a;
}

// store a v8f accumulator (8 consecutive VGPRs) as two ds_store_b128
static __device__ __forceinline__ void store8(float* p, v8f v) {
  v4f lo = {v[0], v[1], v[2], v[3]};
  v4f hi = {v[4], v[5], v[6], v[7]};
  *(v4f*)p = lo;
  *(v4f*)(p + 4) = hi;
}

// exact-accuracy gelu: erf via Abramowitz-Stegun 7.1.26 (|err| <= ~2e-7 abs),
// fully branchless: rcp + 5-term Horner + hardware exp + copysign
static __device__ __forceinline__ float gelu_exact(float f) {
  const float z = f * 0.70710678118654752f;
  const float az = fabsf(z);
  const float t = __builtin_amdgcn_rcpf(fmaf(0.3275911f, az, 1.0f));
  float p = fmaf(1.061405429f, t, -1.453152027f);
  p = fmaf(p, t, 1.421413741f);
  p = fmaf(p, t, -0.284496736f);
  p = fmaf(p, t, 0.254829592f);
  p = p * t;
  const float ex = __expf(-az * az);
  const float erfa = fmaf(-p, ex, 1.0f);        // erf(|z|)
  const float erfv = __builtin_copysignf(erfa, z);
  const float hf2 = 0.5f * f;
  return fmaf(hf2, erfv, hf2);                  // 0.5*f*(1+erf(z))
}

__global__ void __launch_bounds__(32 * WAVES) fused_transformer_block(
    const float* __restrict__ x,
    const float* __restrict__ Wk, const float* __restrict__ Wq,
    const float* __restrict__ Wv,
    const float* __restrict__ g1, const float* __restrict__ bb1,
    const float* __restrict__ g2, const float* __restrict__ bb2,
    const float* __restrict__ W1, const float* __restrict__ b1,
    const float* __restrict__ W2, const float* __restrict__ b2,
    float* __restrict__ out, int nblocks) {
  extern __shared__ float smem[];
  const int lane = threadIdx.x & 31;
  const int wave = threadIdx.x >> 5;
  const int l16 = lane & 15;
  const int hf = lane >> 4;                  // 0/1 half of wave
  const int blk = blockIdx.x * WAVES + wave; // wave-uniform => EXEC stays full
  if (blk >= nblocks) return;

  // per-wave private LDS regions, all matrices column-major
  float* R0 = smem + wave * PW;   // stage-2 x' mirror (16 cols x CS)
  float* STR = R0 + 1088;         // attn strip, 64 cols x SS
  float* RQ = R0 + 2368;          // q; later gelu (32 cols x CS, spans RQ+RK)
  float* RK = R0 + 3456;          // k
  float* RV = R0 + 4544;          // v
  float* GEL = RQ;

  const size_t base = (size_t)blk * (64 * 16);

  const v2f jf = {0.0625f, 0.0625f};  // ones/16 (stats, used as A operand)
  const v2f onef = {1.0f, 1.0f};      // ones (softmax sums / bias chains)

  // causal-mask multiplicative selectors for the diagonal tiles (ti-invariant)
  float msel[8];
#pragma unroll
  for (int r = 0; r < 8; ++r) msel[r] = (l16 > r + 8 * hf) ? 0.0f : 1.0f;

  // ---- per-lane channel-side LN1 params (k = 4c + r2 + 2hf), contiguous b64
  v2f gk1[4], bk1[4];
#pragma unroll
  for (int c = 0; c < 4; ++c) {
    gk1[c] = *(const v2f*)&g1[4 * c + 2 * hf];
    bk1[c] = *(const v2f*)&bb1[4 * c + 2 * hf];
  }

  // ---- QKV weight B-fragments
  v2f wqf[4], wkf[4], wvf[4];
#pragma unroll
  for (int c = 0; c < 4; ++c)
#pragma unroll
    for (int r2 = 0; r2 < 2; ++r2) {
      int k = 4 * c + r2 + 2 * hf;
      wqf[c][r2] = Wq[k * 16 + l16] * 0.25f;  // fold 1/sqrt(C)
      wkf[c][r2] = Wk[k * 16 + l16];
      wvf[c][r2] = Wv[k * 16 + l16];
    }

  // ---- x in D layout (for residuals), kept as v8f per tile
  v8f xr[4];
#pragma unroll
  for (int t = 0; t < 4; ++t)
#pragma unroll
    for (int r = 0; r < 8; ++r)
      xr[t][r] = x[base + (16 * t + r + 8 * hf) * 16 + l16];

  // ---- stage 1 per tile: x A-frags straight from global (contiguous b64),
  //      transposed stats (jf as A operand -> per-lane row stats),
  //      h built as A-fragments in registers, projections -> q,k,v
#pragma unroll
  for (int t = 0; t < 4; ++t) {
    v2f xa[4];
#pragma unroll
    for (int c = 0; c < 4; ++c)
      xa[c] = *(const v2f*)&x[base + (16 * t + l16) * 16 + 4 * c + 2 * hf];

    v8f muT = {}, m2T = {};
#pragma unroll
    for (int c = 0; c < 4; ++c) {
      muT = wmma4(jf, xa[c], muT);          // (J/16) @ X^T : col n = mean(row n)
      m2T = wmma4(jf, xa[c] * xa[c], m2T);  // squares share the fragment
    }
    float mu = muT[0];                       // all 8 regs identical
    float inv = __builtin_amdgcn_rsqf(m2T[0] - mu * mu + EPSV);

    v8f aq = {}, ak = {}, av = {};
#pragma unroll
    for (int c = 0; c < 4; ++c) {
      v2f ha;
#pragma unroll
      for (int r2 = 0; r2 < 2; ++r2)
        ha[r2] = (xa[c][r2] - mu) * inv * gk1[c][r2] + bk1[c][r2];
      aq = wmma4(ha, wqf[c], aq);
      ak = wmma4(ha, wkf[c], ak);
      av = wmma4(ha, wvf[c], av);
    }
    store8(&RQ[l16 * CS + 16 * t + 8 * hf], aq);
    store8(&RK[l16 * CS + 16 * t + 8 * hf], ak);
    store8(&RV[l16 * CS + 16 * t + 8 * hf], av);
  }

  // ---- hoist all k and v fragments into registers (defeats alias re-loads)
  v2f kf[4][4];
#pragma unroll
  for (int si = 0; si < 4; ++si)
#pragma unroll
    for (int c = 0; c < 4; ++c)
      kf[si][c] = ldpair(RK, (4 * c + 2 * hf) * CS + 16 * si + l16, CS);
  v2f vf[16];
#pragma unroll
  for (int kc = 0; kc < 16; ++kc)
    vf[kc] = *(const v2f*)&RV[l16 * CS + 4 * kc + 2 * hf];  // contiguous b64

  // ---- causal attention; softmax without max-shift, sums via ones-WMMA
#pragma unroll
  for (int ti = 0; ti < 4; ++ti) {
    v2f qf[4];
#pragma unroll
    for (int c = 0; c < 4; ++c)
      qf[c] = ldpair(RQ, (4 * c + 2 * hf) * CS + 16 * ti + l16, CS);
#pragma unroll
    for (int si = 0; si <= ti; ++si) {
      v8f acc = {};
#pragma unroll
      for (int c = 0; c < 4; ++c) acc = wmma4(qf[c], kf[si][c], acc);
      v8f pp;
#pragma unroll
      for (int r = 0; r < 8; ++r) {
        float p = __expf(acc[r]);
        if (si == ti) p *= msel[r];           // causal mask (diag tile)
        pp[r] = p;
      }
      store8(&STR[(16 * si + l16) * SS + 8 * hf], pp);  // unnormalized
    }
    v8f y = {}, sm = {};
#pragma unroll
    for (int kc = 0; kc < 4 * (ti + 1); ++kc) {
      v2f a = ldpair(STR, (4 * kc + 2 * hf) * SS + l16, SS);
      y = wmma4(a, vf[kc], y);
      sm = wmma4(a, onef, sm);               // row sums of p
    }
#pragma unroll
    for (int r = 0; r < 8; ++r)
      xr[ti][r] += y[r] * __builtin_amdgcn_rcpf(sm[r]);  // normalize + residual
  }

  // ---- stage 2: mirror x' (only remaining transpose bounce for LN),
  //      transposed stats, h2 as A-frags in registers, FFN1 + gelu
  v2f gk2[4], bk2[4];
#pragma unroll
  for (int c = 0; c < 4; ++c) {
    gk2[c] = *(const v2f*)&g2[4 * c + 2 * hf];
    bk2[c] = *(const v2f*)&bb2[4 * c + 2 * hf];
  }
  v2f w1f[2][4];
#pragma unroll
  for (int g = 0; g < 2; ++g)
#pragma unroll
    for (int c = 0; c < 4; ++c)
#pragma unroll
      for (int r2 = 0; r2 < 2; ++r2)
        w1f[g][c][r2] = W1[(4 * c + r2 + 2 * hf) * 32 + 16 * g + l16];
  v2f bf1[2];
#pragma unroll
  for (int g = 0; g < 2; ++g) {
    float bv = b1[16 * g + l16] * 0.25f;  // ones(16x4) @ (b/4) = b
    bf1[g][0] = bv;
    bf1[g][1] = bv;
  }

#pragma unroll
  for (int t = 0; t < 4; ++t) {
    store8(&R0[l16 * CS + 16 * t + 8 * hf], xr[t]);  // x' mirror (no repack)

    v2f xa[4];
#pragma unroll
    for (int c = 0; c < 4; ++c)
      xa[c] = ldpair(R0, (4 * c + 2 * hf) * CS + 16 * t + l16, CS);

    v8f muT = {}, m2T = {};
#pragma unroll
    for (int c = 0; c < 4; ++c) {
      muT = wmma4(jf, xa[c], muT);
      m2T = wmma4(jf, xa[c] * xa[c], m2T);
    }
    float mu = muT[0];
    float inv = __builtin_amdgcn_rsqf(m2T[0] - mu * mu + EPSV);

    v2f ha[4];
#pragma unroll
    for (int c = 0; c < 4; ++c)
#pragma unroll
      for (int r2 = 0; r2 < 2; ++r2)
        ha[c][r2] = (xa[c][r2] - mu) * inv * gk2[c][r2] + bk2[c][r2];

#pragma unroll
    for (int g = 0; g < 2; ++g) {
      v8f z = {};
      v8f acc = wmma4(onef, bf1[g], z);  // bias via ones-WMMA (C = inline 0)
#pragma unroll
      for (int c = 0; c < 4; ++c) acc = wmma4(ha[c], w1f[g][c], acc);
      v8f gv;
#pragma unroll
      for (int r = 0; r < 8; ++r) gv[r] = gelu_exact(acc[r]);
      store8(&GEL[(16 * g + l16) * CS + 16 * t + 8 * hf], gv);
    }
  }

  // ---- FFN layer 2 (32->16) + residual -> out
  v2f w2f[8];
#pragma unroll
  for (int c = 0; c < 8; ++c)
#pragma unroll
    for (int r2 = 0; r2 < 2; ++r2)
      w2f[c][r2] = W2[(4 * c + r2 + 2 * hf) * 16 + l16];
  v2f bf2;
  bf2[0] = bf2[1] = b2[l16] * 0.25f;

#pragma unroll
  for (int t = 0; t < 4; ++t) {
    v8f z = {};
    v8f acc = wmma4(onef, bf2, z);  // bias via ones-WMMA
#pragma unroll
    for (int c = 0; c < 8; ++c) {
      v2f a = ldpair(GEL, (4 * c + 2 * hf) * CS + 16 * t + l16, CS);
      acc = wmma4(a, w2f[c], acc);
    }
#pragma unroll
    for (int r = 0; r < 8; ++r) {
      int row = 16 * t + r + 8 * hf;
      out[base + row * 16 + l16] = xr[t][r] + acc[r];
    }
  }
}

extern "C" void kernel_launch(void* const* d_in, const int* in_sizes, int n_in,
                              void* d_out, int out_size, void* d_ws,
                              size_t ws_size, hipStream_t stream) {
  (void)n_in; (void)d_ws; (void)ws_size; (void)out_size;
  const float* x   = (const float*)d_in[0];
  const float* Wk  = (const float*)d_in[1];
  const float* Wq  = (const float*)d_in[2];
  const float* Wv  = (const float*)d_in[3];
  const float* g1  = (const float*)d_in[4];
  const float* bb1 = (const float*)d_in[5];
  const float* g2  = (const float*)d_in[6];
  const float* bb2 = (const float*)d_in[7];
  const float* W1  = (const float*)d_in[8];
  const float* b1  = (const float*)d_in[9];
  const float* W2  = (const float*)d_in[10];
  const float* b2  = (const float*)d_in[11];
  float* out = (float*)d_out;

  int nblocks = in_sizes[0] / (64 * 16);  // 16384
  dim3 grid((nblocks + WAVES - 1) / WAVES);
  dim3 block(32 * WAVES);
  size_t shmem = (size_t)WAVES * PW * sizeof(float);
  hipLaunchKernelGGL(fused_transformer_block, grid, block, shmem, stream, x, Wk,
                     Wq, Wv, g1, bb1, g2, bb2, W1, b1, W2, b2, out, nblocks);
}